// NTM_44796508897646
// MI455X (gfx1250) — compile-verified
//
#include <hip/hip_runtime.h>
#include <hip/hip_bf16.h>
#include <math.h>

#define BB 32
#define NSLOTS 16384
#define MM 64
#define HID 512
#define COO 256
#define HPP 70

typedef __attribute__((ext_vector_type(2))) float v2f;
typedef __attribute__((ext_vector_type(8))) float v8f;

// ---------------- workspace layout (float offsets) ----------------
#define OFF_INP    0u          // 32*128
#define OFF_G      4096u       // 32*2048 (reused for both LSTM layers)
#define OFF_H1     69632u      // 32*512
#define OFF_H2     86016u      // 32*512
#define OFF_CTRL   102400u     // 32*256
#define OFF_OHEAD  110592u     // [head][b][70] = 4480
#define OFF_EARAW  115072u     // 32*128
#define OFF_KEY    119168u     // [head][b][64] = 4096
#define OFF_KNORM  123264u     // [head*32+b] 64
#define OFF_STREN  123328u     // 64
#define OFF_GATE   123392u     // 64
#define OFF_SHARP  123456u     // 64
#define OFF_SHIFT  123520u     // [head*32+b][3] 192
#define OFF_E      123712u     // 32*64
#define OFF_A      125760u     // 32*64
#define OFF_ROUT   127808u     // 32*64
#define OFF_ROWS   129856u     // [head*32+b] wt row sums, 64
#define OFF_SUMWT  129920u     // 2
#define OFF_S      131072u     // [head][b][16384] = 1048576 (scores, then wt in place)
#define OFF_PART   1179648u    // [b][chunk(64)][m(64)] = 131072

__device__ __forceinline__ float sigf(float x){ return 1.f/(1.f+expf(-x)); }
__device__ __forceinline__ float softplusf(float x){ return fmaxf(x,0.f)+log1pf(expf(-fabsf(x))); }

__device__ __forceinline__ float waveRed(float v, bool mx){
  #pragma unroll
  for (int o=16;o>0;o>>=1){ float t=__shfl_down(v,o,32); v = mx? fmaxf(v,t) : v+t; }
  return v;
}
// red: 32-float LDS scratch. Leading syncthreads fences any prior LDS use;
// trailing syncthreads fences before caller reuses the scratch region.
__device__ float blockRed1024(float v, float* red, bool mx){
  v = waveRed(v, mx);
  int wid = threadIdx.x>>5, lid = threadIdx.x&31;
  __syncthreads();                 // protect red (and any aliased LDS) from previous use
  if (lid==0) red[wid]=v;
  __syncthreads();
  float r = mx ? -3.4e38f : 0.f;
  if (threadIdx.x < 32) r = red[threadIdx.x];
  if (wid==0){ r = waveRed(r, mx); if (lid==0) red[0]=r; }
  __syncthreads();
  float out = red[0];
  __syncthreads();
  return out;
}

// ---------------- concat [x | prev_read] -> inp [32,128] ----------------
__global__ __launch_bounds__(256) void k_concat(const float* __restrict__ x,
                                                const float* __restrict__ pr,
                                                float* __restrict__ inp){
  int id = blockIdx.x*256 + threadIdx.x;
  if (id >= BB*128) return;
  int b = id>>7, j = id&127;
  inp[id] = (j < 64) ? x[b*64+j] : pr[b*64 + (j-64)];
}

// ---------------- f32 WMMA GEMM: out[32,Nout] = A1@W1^T (+A2@W2^T) + b1 (+b2) ----------------
// W stored [Nout,K] row-major (torch layout); B-fragment B[k][n] = W[n][k].
__global__ __launch_bounds__(32) void k_gemm_wmma(
    const float* __restrict__ A1, int K1, const float* __restrict__ W1,
    const float* __restrict__ A2, int K2, const float* __restrict__ W2,
    const float* __restrict__ b1, const float* __restrict__ b2,
    float* __restrict__ out, int Nout, int act)
{
  const int nTilesN  = Nout >> 4;
  const int rowBase  = (blockIdx.x / nTilesN) << 4;
  const int colBase  = (blockIdx.x % nTilesN) << 4;
  const int lane = threadIdx.x;
  const int hi = lane >> 4;      // half-lane group selects K pair {0,1} vs {2,3}
  const int l  = lane & 15;
  v8f c = {0.f,0.f,0.f,0.f,0.f,0.f,0.f,0.f};
  {
    const float* ar = A1 + (size_t)(rowBase + l) * K1 + 2*hi;
    const float* br = W1 + (size_t)(colBase + l) * K1 + 2*hi;
    for (int k = 0; k < K1; k += 4) {
      v2f a, bfr;
      a.x = ar[k];  a.y = ar[k+1];
      bfr.x = br[k]; bfr.y = br[k+1];
      c = __builtin_amdgcn_wmma_f32_16x16x4_f32(false, a, false, bfr, (short)0, c, false, false);
    }
  }
  if (A2) {
    const float* ar = A2 + (size_t)(rowBase + l) * K2 + 2*hi;
    const float* br = W2 + (size_t)(colBase + l) * K2 + 2*hi;
    for (int k = 0; k < K2; k += 4) {
      v2f a, bfr;
      a.x = ar[k];  a.y = ar[k+1];
      bfr.x = br[k]; bfr.y = br[k+1];
      c = __builtin_amdgcn_wmma_f32_16x16x4_f32(false, a, false, bfr, (short)0, c, false, false);
    }
  }
  float bias = 0.f;
  if (b1) bias += b1[colBase + l];
  if (b2) bias += b2[colBase + l];
  #pragma unroll
  for (int v = 0; v < 8; ++v) {
    int mrow = v + 8*hi;                       // C/D layout: lane<16 -> M=v, lane>=16 -> M=v+8, N=l
    float val = c[v] + bias;
    if (act == 1) val = sigf(val);
    out[(size_t)(rowBase + mrow) * Nout + colBase + l] = val;
  }
}

// ---------------- LSTM gating: g[32,2048] (i,f,g,o) + c_prev -> h ----------------
__global__ __launch_bounds__(256) void k_lstm_gate(const float* __restrict__ g,
                                                   const float* __restrict__ cprev,
                                                   float* __restrict__ hout){
  int id = blockIdx.x*256 + threadIdx.x;
  if (id >= BB*HID) return;
  int b = id>>9, h = id&511;
  const float* gr = g + (size_t)b*2048;
  float gi = gr[h], gf = gr[512+h], gg = gr[1024+h], go = gr[1536+h];
  float c2 = sigf(gf)*cprev[id] + sigf(gi)*tanhf(gg);
  hout[id] = sigf(go)*tanhf(c2);
}

// ---------------- head fcs + erase/add fc (raw) ----------------
__global__ __launch_bounds__(256) void k_headfc(const float* __restrict__ ctrl,
    const float* __restrict__ Wr, const float* __restrict__ br_,
    const float* __restrict__ Ww, const float* __restrict__ bw_,
    const float* __restrict__ Wea, const float* __restrict__ bea,
    float* __restrict__ oHead, float* __restrict__ eaRaw){
  int id = blockIdx.x*256 + threadIdx.x;
  const int NH = 2*BB*HPP;               // 4480
  if (id >= NH + BB*2*MM) return;        // 8576
  if (id < NH) {
    int head = id / (BB*HPP); int r = id % (BB*HPP); int b = r/HPP, j = r%HPP;
    const float* wrow = (head ? Ww : Wr) + (size_t)j*COO;
    const float* crow = ctrl + (size_t)b*COO;
    float acc = (head ? bw_ : br_)[j];
    for (int k=0;k<COO;++k) acc += crow[k]*wrow[k];
    oHead[(size_t)head*BB*HPP + b*HPP + j] = acc;
  } else {
    int r = id - NH; int b = r/(2*MM), j = r%(2*MM);
    const float* wrow = Wea + (size_t)j*COO;
    const float* crow = ctrl + (size_t)b*COO;
    float acc = bea[j];
    for (int k=0;k<COO;++k) acc += crow[k]*wrow[k];
    eaRaw[b*2*MM + j] = acc;
  }
}

// ---------------- per-(head,b) head params; e/a ----------------
__global__ __launch_bounds__(256) void k_params(const float* __restrict__ oHead,
    const float* __restrict__ eaRaw, float* __restrict__ key, float* __restrict__ knorm,
    float* __restrict__ stren, float* __restrict__ gate, float* __restrict__ sharp,
    float* __restrict__ shift, float* __restrict__ eA, float* __restrict__ aA){
  int t = threadIdx.x;
  if (t < 64) {                          // t = head*32 + b
    const float* o = oHead + (size_t)t*HPP;
    float* kp = key + (size_t)t*64;
    float ss = 0.f;
    for (int i=0;i<64;++i){ float kv = tanhf(o[i]); kp[i]=kv; ss += kv*kv; }
    knorm[t] = sqrtf(ss);
    stren[t] = softplusf(o[64]);
    gate[t]  = sigf(o[65]);
    sharp[t] = 1.f + softplusf(o[66]);
    float s0=o[67], s1=o[68], s2=o[69];
    float mx = fmaxf(s0, fmaxf(s1,s2));
    float e0=expf(s0-mx), e1=expf(s1-mx), e2=expf(s2-mx);
    float inv = 1.f/(e0+e1+e2);
    shift[t*3+0]=e0*inv; shift[t*3+1]=e1*inv; shift[t*3+2]=e2*inv;
  } else if (t < 96) {
    int b = t-64;
    for (int i=0;i<64;++i){
      eA[b*64+i] = sigf(eaRaw[b*128 + i]);
      aA[b*64+i] = tanhf(eaRaw[b*128 + 64 + i]);
    }
  }
}

// ---------------- pass 1: content-addressing scores, both heads ----------------
__global__ __launch_bounds__(256) void k_scores(const float* __restrict__ mem,
    const float* __restrict__ key, const float* __restrict__ knorm,
    const float* __restrict__ stren, float* __restrict__ sArr){
  __shared__ __align__(16) float kr[64];
  __shared__ __align__(16) float kw[64];
  int b = blockIdx.y, t = threadIdx.x;
  if (t < 64) kr[t] = key[b*64 + t];
  else if (t < 128) kw[t-64] = key[(size_t)BB*64 + b*64 + (t-64)];
  __syncthreads();
  int n = blockIdx.x*256 + t;
  const float4* row = (const float4*)(mem + ((size_t)b*NSLOTS + n)*MM);
  const float4* k4r = (const float4*)kr;
  const float4* k4w = (const float4*)kw;
  float dr=0.f, dw=0.f, ss=0.f;
  #pragma unroll
  for (int i=0;i<16;++i){
    float4 v = row[i], a = k4r[i], c = k4w[i];
    dr += v.x*a.x + v.y*a.y + v.z*a.z + v.w*a.w;
    dw += v.x*c.x + v.y*c.y + v.z*c.z + v.w*c.w;
    ss += v.x*v.x + v.y*v.y + v.z*v.z + v.w*v.w;
  }
  float nm = sqrtf(ss);
  float denr = fmaxf(nm*knorm[b],    1e-16f);
  float denw = fmaxf(nm*knorm[32+b], 1e-16f);
  sArr[(size_t)b*NSLOTS + n]                       = stren[b]   *dr/denr;
  sArr[(size_t)BB*NSLOTS + (size_t)b*NSLOTS + n]   = stren[32+b]*dw/denw;
}

// ---------------- softmax -> interpolate -> circular shift -> pow ----------------
// Exactly 64KB dynamic LDS: wg[16384]. wg[0:32] doubles as reduction scratch,
// used strictly before wg is populated / after wg is fully consumed (fenced by
// blockRed1024's leading/trailing __syncthreads). Softmax stages live in
// registers (16 elements per thread).
__global__ __launch_bounds__(1024) void k_softshift(float* __restrict__ sArr,
    const float* __restrict__ wprevR, const float* __restrict__ wprevW,
    const float* __restrict__ gateA, const float* __restrict__ sharpA,
    const float* __restrict__ shiftA, float* __restrict__ rowSum){
  extern __shared__ float wg[];   // NSLOTS floats
  int b = blockIdx.x, head = blockIdx.y, t = threadIdx.x;
  float* s = sArr + (size_t)head*BB*NSLOTS + (size_t)b*NSLOTS;
  const float* wprev = (head ? wprevW : wprevR) + (size_t)b*NSLOTS;

  float v[16];
  float lm = -3.4e38f;
  #pragma unroll
  for (int j=0;j<16;++j){ v[j] = s[t + j*1024]; lm = fmaxf(lm, v[j]); }
  float mx = blockRed1024(lm, wg, true);
  float ls = 0.f;
  #pragma unroll
  for (int j=0;j<16;++j){ v[j] = expf(v[j]-mx); ls += v[j]; }
  float S = blockRed1024(ls, wg, false);
  float invS = 1.f/S;
  float g = gateA[head*32+b];
  // blockRed1024 ended with a barrier: safe to populate wg now
  #pragma unroll
  for (int j=0;j<16;++j){
    int i = t + j*1024;
    wg[i] = g*(v[j]*invS) + (1.f-g)*wprev[i];
  }
  __syncthreads();
  float sh0 = shiftA[(head*32+b)*3+0];
  float sh1 = shiftA[(head*32+b)*3+1];
  float sh2 = shiftA[(head*32+b)*3+2];
  float sp  = sharpA[head*32+b];
  float lw = 0.f;
  #pragma unroll
  for (int j=0;j<16;++j){
    int i = t + j*1024;
    float val = wg[(i+NSLOTS-1)&(NSLOTS-1)]*sh0 + wg[i]*sh1 + wg[(i+1)&(NSLOTS-1)]*sh2;
    float wt = powf(val, sp);
    s[i] = wt;                  // scores buffer reused for wt, in place
    lw += wt;
  }
  float T = blockRed1024(lw, wg, false);  // leading barrier fences the wg reads above
  if (t==0) rowSum[head*32+b] = T;
}

// ---------------- deterministic global wt sums ----------------
__global__ __launch_bounds__(64) void k_sumwt(const float* __restrict__ rowSum,
                                              float* __restrict__ sumWT){
  int t = threadIdx.x;
  if (t < 2) {
    float s = 0.f;
    for (int b=0;b<BB;++b) s += rowSum[t*32+b];
    sumWT[t] = s;
  }
}

// ---------------- pass 2: mem_new + read partials ----------------
__global__ __launch_bounds__(256) void k_apply(const float* __restrict__ mem,
    const float* __restrict__ sArr, const float* __restrict__ sumWT,
    const float* __restrict__ eA, const float* __restrict__ aA,
    float* __restrict__ memNew, float* __restrict__ part){
  __shared__ float wrS[256], wwS[256], prS[256];
  int chunk = blockIdx.x, b = blockIdx.y, t = threadIdx.x;
  int n = chunk*256 + t;
  float invR = 1.f/(sumWT[0]+1e-16f);
  float invW = 1.f/(sumWT[1]+1e-16f);
  wrS[t] = sArr[(size_t)b*NSLOTS + n]*invR;
  wwS[t] = sArr[(size_t)BB*NSLOTS + (size_t)b*NSLOTS + n]*invW;
  __syncthreads();
  int m = t & 63, grp = t >> 6;
  float ev = eA[b*64+m], av = aA[b*64+m];
  const float* mb = mem    + ((size_t)b*NSLOTS + (size_t)chunk*256)*MM;
  float*       ob = memNew + ((size_t)b*NSLOTS + (size_t)chunk*256)*MM;
  float acc = 0.f;
  int rbase = grp*64;
  for (int r=0;r<64;++r){
    int row = rbase + r;
    float mv = mb[(size_t)row*MM + m];
    float ww = wwS[row];
    ob[(size_t)row*MM + m] = mv*(1.f - ww*ev) + ww*av;
    acc += wrS[row]*mv;
  }
  prS[t] = acc;
  __syncthreads();
  if (t < 64) part[((size_t)b*64 + chunk)*64 + t] = prS[t]+prS[64+t]+prS[128+t]+prS[192+t];
}

// ---------------- reduce read partials -> read_out [32,64] ----------------
__global__ __launch_bounds__(256) void k_redread(const float* __restrict__ part,
                                                 float* __restrict__ rout){
  int id = blockIdx.x*256 + threadIdx.x;
  if (id >= BB*MM) return;
  int b = id>>6, m = id&63;
  float s = 0.f;
  for (int c=0;c<64;++c) s += part[((size_t)b*64 + c)*64 + m];
  rout[id] = s;
}

// ---------------- output fc: sigmoid([x|read_out] @ W_out^T + b) ----------------
__global__ __launch_bounds__(256) void k_outfc(const float* __restrict__ x,
    const float* __restrict__ rout, const float* __restrict__ W,
    const float* __restrict__ bias, float* __restrict__ out){
  int id = blockIdx.x*256 + threadIdx.x;
  if (id >= BB*64) return;
  int b = id>>6, o = id&63;
  const float* wrow = W + (size_t)o*128;
  const float* xr = x + (size_t)b*64;
  const float* rr = rout + (size_t)b*64;
  float acc = bias[o];
  for (int k=0;k<64;++k) acc += xr[k]*wrow[k];
  for (int k=0;k<64;++k) acc += rr[k]*wrow[64+k];
  out[id] = sigf(acc);
}

extern "C" void kernel_launch(void* const* d_in, const int* in_sizes, int n_in,
                              void* d_out, int out_size, void* d_ws, size_t ws_size,
                              hipStream_t stream) {
  const float* x        = (const float*)d_in[0];
  const float* prevread = (const float*)d_in[1];
  const float* memory   = (const float*)d_in[2];
  const float* h0       = (const float*)d_in[3];   // [2,32,512]
  const float* c0       = (const float*)d_in[4];   // [2,32,512]
  const float* wpr      = (const float*)d_in[5];
  const float* wpw      = (const float*)d_in[6];
  const float* W_ih0    = (const float*)d_in[7];
  const float* W_hh0    = (const float*)d_in[8];
  const float* b_ih0    = (const float*)d_in[9];
  const float* b_hh0    = (const float*)d_in[10];
  const float* W_ih1    = (const float*)d_in[11];
  const float* W_hh1    = (const float*)d_in[12];
  const float* b_ih1    = (const float*)d_in[13];
  const float* b_hh1    = (const float*)d_in[14];
  const float* W_ctr    = (const float*)d_in[15];
  const float* b_ctr    = (const float*)d_in[16];
  const float* W_read   = (const float*)d_in[17];
  const float* b_read   = (const float*)d_in[18];
  const float* W_write  = (const float*)d_in[19];
  const float* b_write  = (const float*)d_in[20];
  const float* W_ea     = (const float*)d_in[21];
  const float* b_ea     = (const float*)d_in[22];
  const float* W_outw   = (const float*)d_in[23];
  const float* b_out    = (const float*)d_in[24];

  float* out    = (float*)d_out;
  float* memNew = out + BB*64;          // mem_new follows out in return order
  float* w      = (float*)d_ws;

  float* inp   = w + OFF_INP;
  float* g     = w + OFF_G;
  float* h1    = w + OFF_H1;
  float* h2    = w + OFF_H2;
  float* ctrl  = w + OFF_CTRL;
  float* oHead = w + OFF_OHEAD;
  float* eaRaw = w + OFF_EARAW;
  float* key   = w + OFF_KEY;
  float* knorm = w + OFF_KNORM;
  float* stren = w + OFF_STREN;
  float* gate  = w + OFF_GATE;
  float* sharp = w + OFF_SHARP;
  float* shift = w + OFF_SHIFT;
  float* eA    = w + OFF_E;
  float* aA    = w + OFF_A;
  float* rout  = w + OFF_ROUT;
  float* rows  = w + OFF_ROWS;
  float* sumWT = w + OFF_SUMWT;
  float* sArr  = w + OFF_S;
  float* part  = w + OFF_PART;

  // 1. inp = [x | prev_read]
  k_concat<<<16, 256, 0, stream>>>(x, prevread, inp);
  // 2. LSTM layer 0 GEMM (WMMA f32): g = inp@W_ih0^T + h0[0]@W_hh0^T + b
  k_gemm_wmma<<<2*(2048/16), 32, 0, stream>>>(inp, 128, W_ih0, h0, 512, W_hh0,
                                              b_ih0, b_hh0, g, 2048, 0);
  k_lstm_gate<<<64, 256, 0, stream>>>(g, c0, h1);
  // 3. LSTM layer 1
  k_gemm_wmma<<<2*(2048/16), 32, 0, stream>>>(h1, 512, W_ih1, h0 + 32*512, 512, W_hh1,
                                              b_ih1, b_hh1, g, 2048, 0);
  k_lstm_gate<<<64, 256, 0, stream>>>(g, c0 + 32*512, h2);
  // 4. controller output: ctrl = sigmoid(h2@W_ctr^T + b)
  k_gemm_wmma<<<2*(256/16), 32, 0, stream>>>(h2, 512, W_ctr, nullptr, 0, nullptr,
                                             b_ctr, nullptr, ctrl, 256, 1);
  // 5. head fcs + erase/add fc
  k_headfc<<<34, 256, 0, stream>>>(ctrl, W_read, b_read, W_write, b_write,
                                   W_ea, b_ea, oHead, eaRaw);
  k_params<<<1, 256, 0, stream>>>(oHead, eaRaw, key, knorm, stren, gate, sharp,
                                  shift, eA, aA);
  // 6. pass 1 over memory: scores for both heads
  k_scores<<<dim3(64, 32), 256, 0, stream>>>(memory, key, knorm, stren, sArr);
  // 7. softmax / interpolate / shift / sharpen per (b, head) — 64KB LDS exactly
  k_softshift<<<dim3(32, 2), 1024, NSLOTS*sizeof(float), stream>>>(
      sArr, wpr, wpw, gate, sharp, shift, rows);
  k_sumwt<<<1, 64, 0, stream>>>(rows, sumWT);
  // 8. pass 2: mem_new + read partials
  k_apply<<<dim3(64, 32), 256, 0, stream>>>(memory, sArr, sumWT, eA, aA, memNew, part);
  k_redread<<<8, 256, 0, stream>>>(part, rout);
  // 9. output fc
  k_outfc<<<8, 256, 0, stream>>>(x, rout, W_outw, b_out, out);
}